// Embedding_44994077393181
// MI455X (gfx1250) — compile-verified
//
#include <hip/hip_runtime.h>
#include <stdint.h>

// Problem constants (from the reference)
#define EMB_DIM 1024
#define VOCAB   32000
#define CTX     2048
#define BATCH   2
#define THREADS 256
#define FULL_ITERS 31        // 31 * 256 lanes * 4 floats = 31744 elements
#define TAIL_BASE  31744     // + tid (256 lanes) -> 32000

typedef unsigned int u32x4 __attribute__((ext_vector_type(4)));
typedef float        f32x4 __attribute__((ext_vector_type(4)));

// GCC-style vector to exactly match the builtin's parameter pointee type:
//   '__attribute__((__vector_size__(4 * sizeof(int)))) int'
typedef int v4i __attribute__((vector_size(4 * sizeof(int))));
typedef __attribute__((address_space(1))) v4i* g_v4i_ptr;   // global (AS1)
typedef __attribute__((address_space(3))) v4i* l_v4i_ptr;   // LDS (AS3)

// gfx1250 async global->LDS copy (ASYNCcnt path). Guarded so compile never fails.
#if defined(__gfx1250__) && __has_builtin(__builtin_amdgcn_global_load_async_to_lds_b128)
#define USE_ASYNC_LDS 1
#else
#define USE_ASYNC_LDS 0
#endif

__global__ __launch_bounds__(THREADS)
void emb_onehot_gather_kernel(const float* __restrict__ X,
                              const float* __restrict__ W,
                              const float* __restrict__ P,
                              float* __restrict__ out)
{
    const int tid = threadIdx.x;
    const int row = blockIdx.x;          // b * CTX + t, 0..4095
    const int t   = row & (CTX - 1);

    __shared__ int   s_idx;
    __shared__ float s_pos[EMB_DIM];     // 4 KB positional row staged via async DMA

    const float* prow = P + (size_t)t * EMB_DIM;

#if USE_ASYNC_LDS
    // One async b128 per lane: 256 lanes * 16B = the whole 4KB pos row.
    // Overlaps with the 128KB one-hot scan below; tracked by ASYNCcnt.
    __builtin_amdgcn_global_load_async_to_lds_b128(
        (g_v4i_ptr)(uintptr_t)(prow + tid * 4),
        (l_v4i_ptr)(uint32_t)(uintptr_t)(&s_pos[tid * 4]),
        0, 0);
#endif

    if (tid == 0) s_idx = 0;
    __syncthreads();

    // ---- Phase 1: find the hot index in this one-hot row (streaming, NT) ----
    const u32x4* __restrict__ x4 = (const u32x4*)(X + (size_t)row * VOCAB);
#pragma unroll 4
    for (int it = 0; it < FULL_ITERS; ++it) {
        u32x4 v = __builtin_nontemporal_load(&x4[it * THREADS + tid]);
        if (v.x | v.y | v.z | v.w) {
            int j = (it * THREADS + tid) * 4;     // exactly one nonzero per row
            if      (v.y) j += 1;
            else if (v.z) j += 2;
            else if (v.w) j += 3;
            s_idx = j;                            // single writer: race-free
        }
    }
    {   // tail: elements 31744..31999, one scalar per lane
        const unsigned* __restrict__ xu = (const unsigned*)(X + (size_t)row * VOCAB);
        unsigned v = __builtin_nontemporal_load(&xu[TAIL_BASE + tid]);
        if (v) s_idx = TAIL_BASE + tid;
    }
    __syncthreads();

    // ---- Phase 2: gather W row (L2-resident, regular temporal) + pos add ----
    const int idx = s_idx;
    const f32x4* __restrict__ w4p = (const f32x4*)(W + (size_t)idx * EMB_DIM);
    f32x4 w4 = w4p[tid];

#if USE_ASYNC_LDS
#if __has_builtin(__builtin_amdgcn_s_wait_asynccnt)
    __builtin_amdgcn_s_wait_asynccnt(0);
#else
    asm volatile("s_wait_asynccnt 0" ::: "memory");
#endif
    f32x4 p4 = ((const f32x4*)s_pos)[tid];        // each lane reads its own bytes
#else
    f32x4 p4 = ((const f32x4*)prow)[tid];
#endif

    f32x4 o = w4 + p4;
    __builtin_nontemporal_store(o, &((f32x4*)(out + (size_t)row * EMB_DIM))[tid]);
}

extern "C" void kernel_launch(void* const* d_in, const int* in_sizes, int n_in,
                              void* d_out, int out_size, void* d_ws, size_t ws_size,
                              hipStream_t stream) {
    (void)in_sizes; (void)n_in; (void)out_size; (void)d_ws; (void)ws_size;
    const float* X = (const float*)d_in[0];   // (2, 2048, 32000) one-hot fp32
    const float* W = (const float*)d_in[1];   // (32000, 1024) fp32
    const float* P = (const float*)d_in[2];   // (2048, 1024) fp32
    float* out = (float*)d_out;               // (2, 2048, 1024) fp32

    dim3 grid(BATCH * CTX);                   // one block per token
    dim3 block(THREADS);
    emb_onehot_gather_kernel<<<grid, block, 0, stream>>>(X, W, P, out);
}